// DeepQNet_62268435857941
// MI455X (gfx1250) — compile-verified
//
#include <hip/hip_runtime.h>

typedef __attribute__((ext_vector_type(2))) float v2f;
typedef __attribute__((ext_vector_type(8))) float v8f;

#define B 8
#define N 2048
#define H 4
#define E 64
#define NEG_SLOPE 0.2f

// workspace layout (float offsets)
#define OFF_C      0                       // c_src[4], c_tgt[4]
#define OFF_MM     16                      // smax[8] @16, smin[8] @24
#define OFF_M2     64                      // 4x64
#define OFF_SE     512                     // B*E state_emb accumulator
#define OFF_SATT   1024                    // B*H*N = 65536
#define OFF_BASE   (1024 + 65536)          // N*E   = 131072
#define OFF_PART   (OFF_BASE + 131072)     // B*N   = 16384
#define OFF_SCONST (OFF_PART + 16384)      // 8

static __device__ __forceinline__ v8f wmma4(v2f a, v2f b, v8f c) {
    // V_WMMA_F32_16X16X4_F32 : D = A(16x4) x B(4x16) + C
    return __builtin_amdgcn_wmma_f32_16x16x4_f32(false, a, false, b, (short)0, c,
                                                 false, false);
}

// ---------------------------------------------------------------------------
// K0: scalars c_src/c_tgt, per-batch state extrema, M2 = (wg @ Wst^T)/H,
//     zero state_emb accumulator.     1 block x 256 threads
__global__ void prep_kernel(const float* __restrict__ state,
                            const float* __restrict__ W_gat,
                            const float* __restrict__ att,
                            const float* __restrict__ Wst,
                            float* __restrict__ ws) {
    int t = threadIdx.x;
    // zero state_emb accumulator (B*E = 512 floats)
    ws[OFF_SE + t]       = 0.0f;
    ws[OFF_SE + 256 + t] = 0.0f;
    if (t < 8) {
        int h = t & 3;
        int off = (t >= 4) ? E : 0;        // src vs tgt half of att
        float acc = 0.0f;
        for (int k = 0; k < E; ++k)
            acc += W_gat[h * E + k] * att[h * 2 * E + off + k];
        ws[OFF_C + t] = acc;
    }
    // M2[h][e] = (1/H) * sum_k wg[h,k] * Wst[e,k]
    {
        int h = t >> 6, e = t & 63;
        float acc = 0.0f;
        for (int k = 0; k < E; ++k)
            acc += W_gat[h * E + k] * Wst[e * E + k];
        ws[OFF_M2 + t] = acc * (1.0f / (float)H);
    }
    // per-batch max/min of state (upper bound for softmax stabilization)
    __shared__ float rmax[256], rmin[256];
    for (int bb = 0; bb < B; ++bb) {
        float mx = -1e30f, mn = 1e30f;
        for (int j = t; j < N; j += 256) {
            float v = state[bb * N + j];
            mx = fmaxf(mx, v);
            mn = fminf(mn, v);
        }
        rmax[t] = mx; rmin[t] = mn;
        __syncthreads();
        for (int d = 128; d >= 1; d >>= 1) {
            if (t < d) {
                rmax[t] = fmaxf(rmax[t], rmax[t + d]);
                rmin[t] = fminf(rmin[t], rmin[t + d]);
            }
            __syncthreads();
        }
        if (t == 0) {
            ws[OFF_MM + bb]     = rmax[0];
            ws[OFF_MM + 8 + bb] = rmin[0];
        }
        __syncthreads();
    }
}

// ---------------------------------------------------------------------------
// K1: masked-softmax-weighted reduction.  s_att[b,h,i] = sum_j p_j * state[b,j]
// Fixed upper-bound stabilization -> single exp per element, no rescaling.
// 1 wave per row i; 8 waves (256 thr) per block; quad-vectorized mask/state.
__global__ void attn_kernel(const float* __restrict__ state,
                            const unsigned char* __restrict__ adj,
                            const float* __restrict__ ws,
                            float* __restrict__ satt) {
    const int NT8 = N / 8;
    int bid = blockIdx.x;
    int b   = bid / (H * NT8);
    int rem = bid % (H * NT8);
    int h   = rem / NT8;
    int t   = rem % NT8;
    int wave = threadIdx.x >> 5;
    int lane = threadIdx.x & 31;

    __shared__ __align__(16) float sst[N];  // state[b, :]  (8 KB)
    for (int j = threadIdx.x; j < N; j += 256) sst[j] = state[b * N + j];
    __syncthreads();

    const float csrc = ws[OFF_C + h];
    const float ctgt = ws[OFF_C + 4 + h];
    const float smax = ws[OFF_MM + b];
    const float smin = ws[OFF_MM + 8 + b];
    const int   i    = t * 8 + wave;
    const float ai   = sst[i] * csrc;
    // upper bound on any unmasked e (lrelu is monotone)
    float M0 = ai + fmaxf(ctgt * smax, ctgt * smin);
    const float Mx = (M0 > 0.0f) ? M0 : NEG_SLOPE * M0;

    const unsigned int* mrow = (const unsigned int*)(adj + (size_t)i * N);
    const float4* sv4 = (const float4*)sst;

    float sacc = 0.0f, wacc = 0.0f;
    for (int q = lane; q < N / 4; q += 32) {
        __builtin_prefetch((const void*)(mrow + q + 256), 0, 1); // global_prefetch_b8
        unsigned int m4 = mrow[q];
        if (m4 != 0x01010101u) {            // at least one unmasked in this quad
            float4 sj = sv4[q];
#define ELEM(comp, sh)                                                  \
            {                                                           \
                float e0 = fmaf(ctgt, sj.comp, ai);                     \
                float e  = (e0 > 0.0f) ? e0 : NEG_SLOPE * e0;           \
                float p  = __expf(e - Mx);                              \
                p = ((m4 >> sh) & 0xffu) ? 0.0f : p;                    \
                sacc += p;                                              \
                wacc  = fmaf(p, sj.comp, wacc);                         \
            }
            ELEM(x, 0) ELEM(y, 8) ELEM(z, 16) ELEM(w, 24)
#undef ELEM
        }
    }
    // wave32 sum-reduce (diagonal is always unmasked -> sacc > 0)
    for (int d = 16; d >= 1; d >>= 1) {
        sacc += __shfl_xor(sacc, d);
        wacc += __shfl_xor(wacc, d);
    }
    if (lane == 0)
        satt[((size_t)b * H + h) * N + i] = wacc / sacc;
}

// ---------------------------------------------------------------------------
// K2: base = strucEmb @ Ws^T  (2048x64 @ 64x64) via f32 WMMA 16x16x4
// 1 wave (32 thr) per 16x16 output tile; grid = 128 i-tiles * 4 e-tiles
__global__ void base_kernel(const float* __restrict__ strucEmb,
                            const float* __restrict__ Ws,
                            float* __restrict__ base) {
    int tile = blockIdx.x;
    int i0 = (tile >> 2) * 16;
    int e0 = (tile & 3) * 16;
    int lane = threadIdx.x;
    int ln   = lane & 15;
    int half = lane >> 4;

    v8f c = {};
    for (int kk = 0; kk < 16; ++kk) {
        int kA = kk * 4 + half * 2;
        v2f a, bb;
        a.x  = strucEmb[(i0 + ln) * E + kA];
        a.y  = strucEmb[(i0 + ln) * E + kA + 1];
        bb.x = Ws[(e0 + ln) * E + kA];       // B[k][n] = Ws[n][k]
        bb.y = Ws[(e0 + ln) * E + kA + 1];
        c = wmma4(a, bb, c);
    }
    for (int r = 0; r < 8; ++r)
        base[(i0 + r + 8 * half) * E + e0 + ln] = c[r];
}

// ---------------------------------------------------------------------------
// K3: x = relu(base + S@M2); state_emb += colsum(x);
//     beta_action = x @ lin2_w^T; part = sum_e relu(ba)*lin3_hi
// block = 128 thr (4 waves), one (b, 16-row i-tile) per block
__global__ void fuse_kernel(const float* __restrict__ satt,
                            const float* __restrict__ M2,
                            const float* __restrict__ base,
                            const float* __restrict__ lin2_w,
                            const float* __restrict__ lin2_b,
                            const float* __restrict__ lin3_w,
                            float* __restrict__ state_emb,
                            float* __restrict__ part) {
    const int TILES = N / 16;              // 128
    int b  = blockIdx.x / TILES;
    int i0 = (blockIdx.x % TILES) * 16;
    int lane = threadIdx.x & 31;
    int wave = threadIdx.x >> 5;
    int ln   = lane & 15;
    int half = lane >> 4;
    int e0   = wave * 16;

    __shared__ float xs[16][65];           // padded: bank = (ln + k) % 64
    __shared__ float rowacc[16];

    // ---- x chunk: K=4 WMMA  (A = s_att 16x4, B = M2 4x16) ----
    v2f a, bb;
    int hA = half * 2;
    a.x  = satt[((size_t)b * H + hA)     * N + i0 + ln];
    a.y  = satt[((size_t)b * H + hA + 1) * N + i0 + ln];
    bb.x = M2[(hA)     * E + e0 + ln];
    bb.y = M2[(hA + 1) * E + e0 + ln];
    v8f c = {};
    c = wmma4(a, bb, c);

    float colsum = 0.0f;
    for (int r = 0; r < 8; ++r) {
        float v = base[(i0 + r + 8 * half) * E + e0 + ln] + c[r];
        v = fmaxf(v, 0.0f);
        xs[r + 8 * half][e0 + ln] = v;
        colsum += v;
    }
    colsum += __shfl_xor(colsum, 16);      // fold two halves of same column
    if (half == 0)
        atomicAdd(&state_emb[b * E + e0 + ln], colsum);

    if (threadIdx.x < 16) rowacc[threadIdx.x] = 0.0f;
    __syncthreads();

    // ---- beta_action chunk: x(16x64) @ lin2_w^T(64x16) via 16 WMMAs ----
    v8f c2 = {};
    for (int kk = 0; kk < 16; ++kk) {
        int kA = kk * 4 + half * 2;
        v2f a2, b2;
        a2.x = xs[ln][kA];
        a2.y = xs[ln][kA + 1];
        b2.x = lin2_w[(e0 + ln) * E + kA];
        b2.y = lin2_w[(e0 + ln) * E + kA + 1];
        c2 = wmma4(a2, b2, c2);
    }
    float l2b = lin2_b[e0 + ln];
    float l3  = lin3_w[E + e0 + ln];       // high half of lin3
    for (int r = 0; r < 8; ++r) {
        float ba = c2[r] + l2b;
        float p  = fmaxf(ba, 0.0f) * l3;
        atomicAdd(&rowacc[r + 8 * half], p);
    }
    __syncthreads();
    if (threadIdx.x < 16)
        part[(size_t)b * N + i0 + threadIdx.x] = rowacc[threadIdx.x];
}

// ---------------------------------------------------------------------------
// K4: beta_state = state_emb @ lin1_w^T + lin1_b;
//     sconst[b] = sum_e relu(bs)*lin3_lo[e] + lin3_b    (1 block x 64 thr)
__global__ void fin_kernel(const float* __restrict__ state_emb,
                           const float* __restrict__ lin1_w,
                           const float* __restrict__ lin1_b,
                           const float* __restrict__ lin3_w,
                           const float* __restrict__ lin3_b,
                           float* __restrict__ sconst) {
    __shared__ float acc[B];
    int e = threadIdx.x;
    if (e < B) acc[e] = 0.0f;
    __syncthreads();
    float l1b = lin1_b[e];
    float l3  = lin3_w[e];                 // low half of lin3
    for (int b = 0; b < B; ++b) {
        float v = l1b;
        for (int k = 0; k < E; ++k)
            v += state_emb[b * E + k] * lin1_w[e * E + k];
        atomicAdd(&acc[b], fmaxf(v, 0.0f) * l3);
    }
    __syncthreads();
    if (e < B) sconst[e] = acc[e] + lin3_b[0];
}

// ---------------------------------------------------------------------------
// K5: out[b,i] = part[b,i] + sconst[b]
__global__ void add_kernel(const float* __restrict__ part,
                           const float* __restrict__ sconst,
                           float* __restrict__ out) {
    int idx = blockIdx.x * blockDim.x + threadIdx.x;
    if (idx < B * N)
        out[idx] = part[idx] + sconst[idx / N];
}

// ---------------------------------------------------------------------------
extern "C" void kernel_launch(void* const* d_in, const int* in_sizes, int n_in,
                              void* d_out, int out_size, void* d_ws, size_t ws_size,
                              hipStream_t stream) {
    const float*         state    = (const float*)d_in[0];
    const float*         strucEmb = (const float*)d_in[1];
    const unsigned char* adj      = (const unsigned char*)d_in[2];
    const float*         W_gat    = (const float*)d_in[3];
    const float*         att      = (const float*)d_in[4];
    const float*         Ws       = (const float*)d_in[5];
    const float*         Wst      = (const float*)d_in[6];
    const float*         lin1_w   = (const float*)d_in[7];
    const float*         lin1_b   = (const float*)d_in[8];
    const float*         lin2_w   = (const float*)d_in[9];
    const float*         lin2_b   = (const float*)d_in[10];
    const float*         lin3_w   = (const float*)d_in[11];
    const float*         lin3_b   = (const float*)d_in[12];
    float* ws  = (float*)d_ws;
    float* out = (float*)d_out;

    prep_kernel<<<1, 256, 0, stream>>>(state, W_gat, att, Wst, ws);
    attn_kernel<<<B * H * (N / 8), 256, 0, stream>>>(state, adj, ws,
                                                     ws + OFF_SATT);
    base_kernel<<<(N / 16) * (E / 16), 32, 0, stream>>>(strucEmb, Ws, ws + OFF_BASE);
    fuse_kernel<<<B * (N / 16), 128, 0, stream>>>(ws + OFF_SATT, ws + OFF_M2,
                                                  ws + OFF_BASE, lin2_w, lin2_b,
                                                  lin3_w, ws + OFF_SE, ws + OFF_PART);
    fin_kernel<<<1, 64, 0, stream>>>(ws + OFF_SE, lin1_w, lin1_b, lin3_w, lin3_b,
                                     ws + OFF_SCONST);
    add_kernel<<<(B * N + 255) / 256, 256, 0, stream>>>(ws + OFF_PART,
                                                        ws + OFF_SCONST, out);
}